// SparseSubMConv3d_55387898249941
// MI455X (gfx1250) — compile-verified
//
#include <hip/hip_runtime.h>
#include <stdint.h>

// ---------------- hash table (rulebook) ----------------
#define TS_LOG 19
#define TSIZE  (1u << TS_LOG)
#define TMASK  (TSIZE - 1u)
#define EMPTY64 0xFFFFFFFFFFFFFFFFull

typedef __attribute__((ext_vector_type(2))) float v2f;
typedef __attribute__((ext_vector_type(8))) float v8f;

__device__ __forceinline__ uint32_t hash_key(uint32_t k) {
    k *= 2654435761u;
    k ^= k >> 15;
    return k & TMASK;
}

__global__ void k_table_init(unsigned long long* __restrict__ table) {
    uint32_t i = blockIdx.x * blockDim.x + threadIdx.x;
    if (i < TSIZE) table[i] = EMPTY64;
}

__global__ void k_table_insert(const int* __restrict__ coords,
                               const int* __restrict__ sshape,
                               unsigned long long* __restrict__ table, int n) {
    int i = blockIdx.x * blockDim.x + threadIdx.x;
    if (i >= n) return;
    int b = coords[i * 4 + 0], x = coords[i * 4 + 1];
    int y = coords[i * 4 + 2], z = coords[i * 4 + 3];
    int X = sshape[0], Y = sshape[1], Z = sshape[2];
    uint32_t key = ((uint32_t)((b * X + x) * Y + y)) * (uint32_t)Z + (uint32_t)z;
    unsigned long long entry = ((unsigned long long)key << 32) | (uint32_t)i;
    uint32_t slot = hash_key(key);
    for (;;) {
        unsigned long long prev = atomicCAS(&table[slot], EMPTY64, entry);
        if (prev == EMPTY64) break;
        slot = (slot + 1u) & TMASK;
    }
}

// ---------------- sparse subm conv via f32 WMMA ----------------
// Block = 256 threads = 8 wave32s.  Block owns 32 output rows.
// Wave w: row-group rg = w>>2 (16 rows), cout tile ct = w&3 (16 cols).
// Per offset k: probe rulebook, stage W[k] (64x64) + gathered feats (32x64)
// in LDS, then 16 x v_wmma_f32_16x16x4_f32 accumulating C in 8 VGPRs.
__launch_bounds__(256)
__global__ void k_subm_conv(const float* __restrict__ feat,
                            const int*   __restrict__ coords,
                            const float* __restrict__ weight,
                            const float* __restrict__ bias,
                            const int*   __restrict__ sshape,
                            const unsigned long long* __restrict__ table,
                            float* __restrict__ out, int n) {
    __shared__ float s_feat[32][68];   // stride 68 dwords: conflict-free A b64 reads
    __shared__ float s_w[64][72];      // stride 72 dwords: half-waves hit disjoint banks
    __shared__ int   s_nbr[32];
    __shared__ int   s_coords[32][4];

    const int t     = threadIdx.x;
    const int lane  = t & 31;
    const int wave  = t >> 5;
    const int rg    = wave >> 2;       // 0..1 row group
    const int ct    = wave & 3;        // 0..3 cout tile
    const int c0    = ct * 16;
    const int l     = lane & 15;
    const int khalf = lane >> 4;       // 0: K pair {0,1} / M 0-7 ; 1: K pair {2,3} / M 8-15

    const int row0 = blockIdx.x * 32;
    const int X = sshape[0], Y = sshape[1], Z = sshape[2];

    if (t < 32) {
        int R = row0 + t;
        int4 c;
        if (R < n) c = ((const int4*)coords)[R];
        else       c = make_int4(-1, -1000000, -1000000, -1000000);
        s_coords[t][0] = c.x; s_coords[t][1] = c.y;
        s_coords[t][2] = c.z; s_coords[t][3] = c.w;
    }
    __syncthreads();

    // C init = bias broadcast down each column
    v8f acc;
    {
        float bv = bias[c0 + l];
#pragma unroll
        for (int i = 0; i < 8; ++i) acc[i] = bv;
    }

    for (int k = 0; k < 27; ++k) {
        __syncthreads();  // previous iteration's LDS consumers are done

        // --- rulebook probe for this offset (32 threads) ---
        if (t < 32) {
            int dx = k / 9 - 1, dy = (k / 3) % 3 - 1, dz = k % 3 - 1;
            int b = s_coords[t][0];
            int x = s_coords[t][1] + dx;
            int y = s_coords[t][2] + dy;
            int z = s_coords[t][3] + dz;
            int nbr = -1;
            if (b >= 0 && x >= 0 && x < X && y >= 0 && y < Y && z >= 0 && z < Z) {
                uint32_t key = ((uint32_t)((b * X + x) * Y + y)) * (uint32_t)Z + (uint32_t)z;
                uint32_t slot = hash_key(key);
                for (;;) {
                    unsigned long long e = table[slot];
                    if (e == EMPTY64) break;
                    if ((uint32_t)(e >> 32) == key) { nbr = (int)(uint32_t)e; break; }
                    slot = (slot + 1u) & TMASK;
                }
            }
            s_nbr[t] = nbr;
        }

        // --- stage W[k] (64x64 f32, 16 dwords / thread) ---
        {
            const float* wk = weight + k * 64 * 64;
            int base = t * 16;
#pragma unroll
            for (int j = 0; j < 4; ++j) {
                int idx = base + j * 4;
                int ci = idx >> 6, co = idx & 63;
                float4 wv = *((const float4*)(wk + idx));
                s_w[ci][co + 0] = wv.x; s_w[ci][co + 1] = wv.y;
                s_w[ci][co + 2] = wv.z; s_w[ci][co + 3] = wv.w;
            }
        }
        __syncthreads();

        // --- gather 32 neighbor rows into LDS (8 threads / row, 32B each) ---
        {
            int row  = t >> 3;
            int part = t & 7;
            int nbr  = s_nbr[row];
            float4 f0 = make_float4(0.f, 0.f, 0.f, 0.f), f1 = f0;
            if (nbr >= 0) {
                const float4* src = (const float4*)(feat + (size_t)nbr * 64 + part * 8);
                f0 = src[0]; f1 = src[1];
            }
            float* dst = &s_feat[row][part * 8];
            ((float4*)dst)[0] = f0;
            ((float4*)dst)[1] = f1;
        }
        __syncthreads();

        // --- 16 k-steps of D = A x B + C on the f32 matrix pipe ---
#pragma unroll
        for (int kk = 0; kk < 16; ++kk) {
            int ci = kk * 4 + khalf * 2;
            v2f a;  // A 16x4: lane l holds rows rg*16+l, K = ci, ci+1
            a.x = s_feat[rg * 16 + l][ci];
            a.y = s_feat[rg * 16 + l][ci + 1];
            v2f bm; // B 4x16: lane l holds col c0+l, K = ci, ci+1
            bm.x = s_w[ci][c0 + l];
            bm.y = s_w[ci + 1][c0 + l];
            acc = __builtin_amdgcn_wmma_f32_16x16x4_f32(
                /*neg_a=*/false, a, /*neg_b=*/false, bm,
                /*c_mod=*/(short)0, acc, /*reuse_a=*/false, /*reuse_b=*/false);
        }
    }

    // --- write 16x16 tile: VGPR i holds row M = khalf*8 + i, col c0+l ---
#pragma unroll
    for (int i = 0; i < 8; ++i) {
        int R = row0 + rg * 16 + khalf * 8 + i;
        if (R < n) out[(size_t)R * 64 + c0 + l] = acc[i];
    }
}

extern "C" void kernel_launch(void* const* d_in, const int* in_sizes, int n_in,
                              void* d_out, int out_size, void* d_ws, size_t ws_size,
                              hipStream_t stream) {
    const float* feat   = (const float*)d_in[0];
    const int*   coords = (const int*)  d_in[1];
    const float* weight = (const float*)d_in[2];
    const float* bias   = (const float*)d_in[3];
    const int*   sshape = (const int*)  d_in[4];
    // d_in[5] = bs (unused; batch comes from coords)

    const int n = in_sizes[1] / 4;   // coords is [N,4]
    unsigned long long* table = (unsigned long long*)d_ws;  // 4 MB of d_ws

    k_table_init  <<<(TSIZE + 255) / 256, 256, 0, stream>>>(table);
    k_table_insert<<<(n + 255) / 256,     256, 0, stream>>>(coords, sshape, table, n);
    k_subm_conv   <<<(n + 31) / 32,       256, 0, stream>>>(feat, coords, weight, bias,
                                                            sshape, table,
                                                            (float*)d_out, n);
}